// EntityAttention_18786186953385
// MI455X (gfx1250) — compile-verified
//
#include <hip/hip_runtime.h>
#include <hip/hip_bf16.h>

// Problem sizes (fixed by the reference)
#define OBS_D  512      // d
#define ATTN_K 128      // k
#define NA     16       // visible agents per entity
#define BE     4096     // 128 batches * 32 entities

typedef __attribute__((ext_vector_type(2))) float v2f;
typedef __attribute__((ext_vector_type(4))) float v4f;
typedef __attribute__((ext_vector_type(8))) float v8f;

#define HAS_WMMA_F32 __has_builtin(__builtin_amdgcn_wmma_f32_16x16x4_f32)

// ---------------------------------------------------------------------------
// Generic wave-tile WMMA GEMM:  C(MxN) = A(MxK) @ B
//   BT == false : B stored (K x N) row-major
//   BT == true  : B stored (N x K) row-major (i.e. we multiply by B^T)
// One wave per 16x16 output tile, V_WMMA_F32_16X16X4_F32 along K.
// ISA layouts (CDNA5 ISA §7.12.2, 32-bit A 16x4 / B 4x16 / C 16x16):
//   A: lane l -> row = l%16, vgpr0 = K(kk + 2*(l/16)), vgpr1 = K(kk + 2*(l/16)+1)
//   B: lane l -> col = l%16, vgpr0 = K(kk + 2*(l/16)), vgpr1 = next K  (mirror of A)
//   C/D: lane l, vgpr v -> M = v + 8*(l/16), N = l%16
// ---------------------------------------------------------------------------
template <int M, int N, int K, bool BT>
__global__ __launch_bounds__(128) void gemm_wmma_f32(const float* __restrict__ A,
                                                     const float* __restrict__ B,
                                                     float* __restrict__ C) {
#if HAS_WMMA_F32
    const int wave = (blockIdx.x * blockDim.x + threadIdx.x) >> 5;
    const int lane = threadIdx.x & 31;
    const int hl   = lane >> 4;   // half-wave select (K split)
    const int r    = lane & 15;   // row/col within tile

    constexpr int ntiles = N / 16;
    const int i0 = (wave / ntiles) * 16;
    const int n0 = (wave % ntiles) * 16;

    v8f acc = {};
    const float* arow = A + (size_t)(i0 + r) * K + 2 * hl;
    const float* brow = BT ? (B + (size_t)(n0 + r) * K + 2 * hl) : nullptr;

    for (int kk = 0; kk < K; kk += 4) {
        v2f a = *(const v2f*)(arow + kk);
        v2f b;
        if (BT) {
            b = *(const v2f*)(brow + kk);                 // contiguous along K
        } else {
            b.x = B[(size_t)(kk + 2 * hl) * N + n0 + r];  // strided along K
            b.y = B[(size_t)(kk + 2 * hl + 1) * N + n0 + r];
        }
        acc = __builtin_amdgcn_wmma_f32_16x16x4_f32(
            /*neg_a=*/false, a, /*neg_b=*/false, b,
            /*c_mod=*/(short)0, acc, /*reuse_a=*/false, /*reuse_b=*/false);
    }

#pragma unroll
    for (int v = 0; v < 8; ++v)
        C[(size_t)(i0 + v + 8 * hl) * N + n0 + r] = acc[v];
#else
    // Scalar fallback (grid-stride so the WMMA grid sizing still covers M*N).
    const int total = M * N;
    for (int idx = blockIdx.x * blockDim.x + threadIdx.x; idx < total;
         idx += gridDim.x * blockDim.x) {
        const int i = idx / N, j = idx % N;
        float s = 0.f;
        for (int k = 0; k < K; ++k)
            s += A[(size_t)i * K + k] * (BT ? B[(size_t)j * K + k]
                                            : B[(size_t)k * N + j]);
        C[idx] = s;
    }
#endif
}

// ---------------------------------------------------------------------------
// Streaming kernel: beta[be,a] = P[be,:] . visible[be,a,:], then stable
// softmax over a (16). One 256-thread block (8 wave32) per (b,e); each wave
// does 2 rows via b128 loads + shuffle reduction. This streams the 134 MB
// `visible` tensor exactly once -> HBM-bandwidth bound (~6 us at 23.3 TB/s).
// ---------------------------------------------------------------------------
__global__ __launch_bounds__(256) void attn_softmax_stream(const float* __restrict__ vis,
                                                           const float* __restrict__ P,
                                                           float* __restrict__ out) {
    const int be   = blockIdx.x;       // 0..BE-1
    const int tid  = threadIdx.x;
    const int wave = tid >> 5;         // 0..7
    const int lane = tid & 31;

    __shared__ float p_sh[OBS_D];
    __shared__ float beta_sh[NA];

    for (int d = tid; d < OBS_D; d += 256)
        p_sh[d] = P[(size_t)be * OBS_D + d];
    __syncthreads();

    const float* vbase = vis + (size_t)be * NA * OBS_D;

#pragma unroll
    for (int aa = 0; aa < 2; ++aa) {
        const int a = wave * 2 + aa;
        const float* vrow = vbase + (size_t)a * OBS_D;
        float s = 0.f;
#pragma unroll
        for (int j = 0; j < 4; ++j) {
            const int d = (j * 32 + lane) * 4;          // 512 = 4 * 32 * 4
            v4f v = *(const v4f*)(vrow + d);            // global_load_b128
            s += v.x * p_sh[d] + v.y * p_sh[d + 1]
               + v.z * p_sh[d + 2] + v.w * p_sh[d + 3];
        }
#pragma unroll
        for (int off = 16; off > 0; off >>= 1)          // wave32 reduction
            s += __shfl_xor(s, off, 32);
        if (lane == 0) beta_sh[a] = s;
    }
    __syncthreads();

    // Stable softmax over the 16 betas (redundantly by 16 threads; trivial).
    if (tid < NA) {
        float m = -INFINITY;
#pragma unroll
        for (int k = 0; k < NA; ++k) m = fmaxf(m, beta_sh[k]);
        float sum = 0.f;
#pragma unroll
        for (int k = 0; k < NA; ++k) sum += __expf(beta_sh[k] - m);
        out[(size_t)be * NA + tid] = __expf(beta_sh[tid] - m) / sum;
    }
}

extern "C" void kernel_launch(void* const* d_in, const int* in_sizes, int n_in,
                              void* d_out, int out_size, void* d_ws, size_t ws_size,
                              hipStream_t stream) {
    (void)in_sizes; (void)n_in; (void)out_size; (void)ws_size;

    const float* agent = (const float*)d_in[0];  // (128,32,512)
    const float* vis   = (const float*)d_in[1];  // (128,32,16,512)
    const float* wpsi  = (const float*)d_in[2];  // (512,128)
    const float* wphi  = (const float*)d_in[3];  // (512,128)
    float* out = (float*)d_out;                  // (128,32,16)

    // Workspace: Q (BE x 128, 2 MB) then P (BE x 512, 8 MB) -> 10 MB total.
    float* Q = (float*)d_ws;
    float* P = Q + (size_t)BE * ATTN_K;

    // 1) Q = Agent @ Wpsi            (4096x512 @ 512x128)
    {
        constexpr int tiles = (BE / 16) * (ATTN_K / 16);   // 2048 waves
        gemm_wmma_f32<BE, ATTN_K, OBS_D, false>
            <<<tiles / 4, 128, 0, stream>>>(agent, wpsi, Q);
    }
    // 2) P = Q @ Wphi^T              (4096x128 @ 128x512, B given as 512x128)
    {
        constexpr int tiles = (BE / 16) * (OBS_D / 16);    // 8192 waves
        gemm_wmma_f32<BE, OBS_D, ATTN_K, true>
            <<<tiles / 4, 128, 0, stream>>>(Q, wphi, P);
    }
    // 3) beta + softmax, streaming `visible` once.
    attn_softmax_stream<<<BE, 256, 0, stream>>>(vis, P, out);
}